// KPConv_MSRes_84739704750343
// MI455X (gfx1250) — compile-verified
//
#include <hip/hip_runtime.h>
#include <hip/hip_bf16.h>

// ---------------------------------------------------------------------------
// KPConv dual-branch block for MI455X (gfx1250, wave32, WMMA f32 16x16x4).
//
// Roofline: ~11 GFLOP total; feature table (10MB) + weights (<1MB) are L2
// resident (192MB), so the kernel is L2-gather / issue bound, not HBM bound.
// f32 WMMA keeps reference numerics at no bandwidth cost. Feature gather is
// staged global->LDS with GLOBAL_LOAD_ASYNC_TO_LDS_B128 (ASYNCcnt tracked).
// ---------------------------------------------------------------------------

#define DEV_INLINE __device__ __forceinline__

constexpr int   N_Q    = 20000;   // query points
constexpr int   M_S    = 20000;   // support points
constexpr int   H_NB   = 32;      // neighbors
constexpr int   C_IN   = 128;     // input channels
constexpr int   D_OUT  = 64;      // per-branch output channels
constexpr int   K_MINI = 7;
constexpr int   K_MID  = 13;
constexpr int   K_TOT  = K_MINI + K_MID;        // 20
constexpr float KP_EXT = 0.6f;
constexpr float EPSN   = 1e-5f;

constexpr int QT = 16;            // queries per block == WMMA M tile
constexpr int QH = 8;             // half-tile for feature staging
static_assert(N_Q % QT == 0, "tile must divide N");
constexpr int NBLK = N_Q / QT;    // 1250

// padded stride for the LDS "weighted" tensor: 20*128=2560 is a multiple of
// 64 banks -> 16-way conflict on the step-2 A-operand reads; +4 fixes it.
constexpr int SW_STRIDE = K_TOT * C_IN + 4;     // 2564
// influence-weight stride: 20 real entries + 1 zero slot so out-of-range
// k-rows clamp to a zero read (no EXEC predication around WMMA operands).
constexpr int KA_STR = K_TOT + 1;               // 21 (gcd(21,64)=1: no conflicts)

typedef __attribute__((ext_vector_type(2))) float v2f;
typedef __attribute__((ext_vector_type(8))) float v8f;

// D = A(16x4) * B(4x16) + C, all f32.  Operand layouts per CDNA5 ISA 7.12.2:
//   A: lanes 0-15 hold {K=0,K=1}, lanes 16-31 hold {K=2,K=3}, M = lane&15
//   B: (symmetric) lanes 0-15 rows {0,1}, lanes 16-31 rows {2,3}, N = lane&15
//   C/D: vgpr j -> row j + (lane>=16 ? 8 : 0), col = lane&15
DEV_INLINE v8f wmma4(float ax, float ay, float bx, float by, v8f c) {
  v2f a = {ax, ay};
  v2f b = {bx, by};
  return __builtin_amdgcn_wmma_f32_16x16x4_f32(false, a, false, b, (short)0, c,
                                               false, false);
}

// Async global->LDS copy, 16B per lane (GV mode, 64-bit vaddr).
// LDS offset = low 32 bits of the generic pointer (ISA 10.2 aperture rule).
DEV_INLINE void async_gather_b128(const float* gsrc, float* ldst) {
  const unsigned lds_off = (unsigned)(uintptr_t)ldst;
  asm volatile("global_load_async_to_lds_b128 %0, %1, off"
               :: "v"(lds_off), "v"(gsrc)
               : "memory");
}
DEV_INLINE void wait_async0() {
  asm volatile("s_wait_asynccnt 0x0" ::: "memory");
}

// dynamic-LDS layout for kernel 1 (bytes, all 16B aligned)
constexpr size_t SM_W   = (size_t)QT * SW_STRIDE * 4;      // 164096 weighted
constexpr size_t SM_NX  = (size_t)QH * H_NB * C_IN * 4;    // 131072 feats [8][32][128]
constexpr size_t SM_A   = (size_t)QH * H_NB * KA_STR * 4;  //  21504 influence [8][32][21]
constexpr size_t SM_IDX = (size_t)QT * H_NB * 4;           //   2048 neighbor indices
constexpr size_t SM_KP  = (size_t)K_TOT * 4 * 4;           //    320 kernel points
constexpr size_t SM_CNT = (size_t)QT * 4;                  //     64 valid counts
constexpr size_t SM_NN  = (size_t)QT * 4;                  //     64 1/n_num
constexpr size_t SMEM1  = SM_W + SM_NX + SM_A + SM_IDX + SM_KP + SM_CNT + SM_NN; // 319168

// ---------------------------------------------------------------------------
// K1: fused dual KPConv. One block = 16 queries, 8 waves.
// ---------------------------------------------------------------------------
__global__ __launch_bounds__(256, 1)
void k1_kpconv_fused(const float* __restrict__ q_pts,
                     const float* __restrict__ s_pts,
                     const int*   __restrict__ nbi,
                     const float* __restrict__ x,
                     const int*   __restrict__ stacklen,
                     const float* __restrict__ kp_mini,
                     const float* __restrict__ w_mini,
                     const float* __restrict__ kp_mid,
                     const float* __restrict__ w_mid,
                     float* __restrict__ x1raw,
                     float* __restrict__ x2raw,
                     float* __restrict__ segsum) {
  extern __shared__ char smem_raw[];
  float* s_w   = (float*)smem_raw;                        // [16][SW_STRIDE]
  float* s_nx  = (float*)(smem_raw + SM_W);               // [8*32][128]
  float* s_a   = (float*)(smem_raw + SM_W + SM_NX);       // [8*32][21]
  int*   s_idx = (int*)  (smem_raw + SM_W + SM_NX + SM_A);          // [16][32]
  float* s_kp  = (float*)(smem_raw + SM_W + SM_NX + SM_A + SM_IDX); // [20][4]
  int*   s_cnt = (int*)  ((char*)s_kp + SM_KP);           // [16]
  float* s_nn  = (float*)((char*)s_cnt + SM_CNT);         // [16]

  const int tid   = threadIdx.x;
  const int lane  = tid & 31;           // wave32
  const int wave  = tid >> 5;           // 8 waves
  const int qbase = blockIdx.x * QT;
  const int len0  = stacklen[0];

  // kernel points (k<7: mini, else mid), counts, neighbor indices
  if (tid < K_TOT) {
    const float* src = (tid < K_MINI) ? (kp_mini + 3 * tid)
                                      : (kp_mid + 3 * (tid - K_MINI));
    s_kp[tid * 4 + 0] = src[0];
    s_kp[tid * 4 + 1] = src[1];
    s_kp[tid * 4 + 2] = src[2];
  }
  if (tid < QT) s_cnt[tid] = 0;
  for (int i = tid; i < QT * H_NB; i += 256)
    s_idx[i] = nbi[qbase * H_NB + i];
  __syncthreads();

  for (int half = 0; half < 2; ++half) {
    // ---- influence weights a[pair][k] (one thread per (q,h) pair) ----
    {
      const int ql = tid >> 5, h = tid & 31;
      const int qi = qbase + half * QH + ql;
      const int idx = s_idx[(half * QH + ql) * H_NB + h];
      float px = 1.0e6f, py = 1.0e6f, pz = 1.0e6f;     // shadow point
      if (idx < M_S) { px = s_pts[idx*3]; py = s_pts[idx*3+1]; pz = s_pts[idx*3+2]; }
      const float rx = px - q_pts[qi*3];
      const float ry = py - q_pts[qi*3+1];
      const float rz = pz - q_pts[qi*3+2];
      float* arow = s_a + tid * KA_STR;      // pair index == tid
      #pragma unroll
      for (int k = 0; k < K_TOT; ++k) {
        float dx = rx - s_kp[k*4], dy = ry - s_kp[k*4+1], dz = rz - s_kp[k*4+2];
        float d2 = dx*dx + dy*dy + dz*dz;
        float dist = sqrtf(fmaxf(d2, 1e-12f));
        arow[k] = fmaxf(1.0f - dist * (1.0f / KP_EXT), 0.0f);
      }
      arow[K_TOT] = 0.0f;                    // clamp slot for padded k-rows
    }
    // ---- feature gather: async global->LDS, 512B per neighbor row ----
    #pragma unroll 1
    for (int pi = 0; pi < 32; ++pi) {
      const int p  = wave * 32 + pi;            // (q,h) pair: q=p>>5, h=p&31
      const int idx = s_idx[half * QH * H_NB + p];
      if (idx < M_S) {                          // wave-uniform branch
        async_gather_b128(x + (size_t)idx * C_IN + lane * 4,
                          s_nx + p * C_IN + lane * 4);
      } else {                                  // shadow feature row = 0
        *(float4*)(s_nx + p * C_IN + lane * 4) = make_float4(0.f,0.f,0.f,0.f);
      }
    }
    wait_async0();
    // ---- density-norm count from the staged rows (coalesced readback) ----
    #pragma unroll 1
    for (int pi = 0; pi < 32; ++pi) {
      const int p = wave * 32 + pi;
      const float4 v = *(const float4*)(s_nx + p * C_IN + lane * 4);
      float fs = v.x + v.y + v.z + v.w;
      #pragma unroll
      for (int off = 16; off > 0; off >>= 1) fs += __shfl_down(fs, off, 32);
      if (lane == 0 && fs > 0.0f) atomicAdd(&s_cnt[half * QH + (p >> 5)], 1);
    }
    __syncthreads();

    // ---- step 1: weighted[k,c] = a^T[k,h] @ nx[h,c], WMMA over h ----
    // 128 sub-tiles this half (8 q x 2 k-tiles x 8 c-subtiles), 16 per wave
    #pragma unroll 1
    for (int t = 0; t < 16; ++t) {
      const int task = wave * 16 + t;
      const int ql   = task >> 4;
      const int tile = (task >> 3) & 1;     // k rows 0..15 / 16..19
      const int csub = task & 7;
      const int qg   = half * QH + ql;
      const int m    = lane & 15;
      const int kk   = tile * 16 + m;       // kernel-point row
      const int kkc  = (kk < K_TOT) ? kk : K_TOT;   // clamp to zero slot
      const int hoff = (lane >> 4) * 2;
      const int ccol = csub * 16 + m;
      const float* anx = s_nx + (ql * H_NB) * C_IN;
      const float* aa  = s_a + (ql * H_NB) * KA_STR;
      // batch all operands, then the 8-deep WMMA accumulate chain
      float axv[8], ayv[8], bxv[8], byv[8];
      #pragma unroll
      for (int ks = 0; ks < H_NB / 4; ++ks) {
        const int hb = ks * 4 + hoff;
        axv[ks] = aa[hb * KA_STR + kkc];
        ayv[ks] = aa[(hb + 1) * KA_STR + kkc];
        bxv[ks] = anx[hb * C_IN + ccol];
        byv[ks] = anx[(hb + 1) * C_IN + ccol];
      }
      v8f acc = {0.f,0.f,0.f,0.f,0.f,0.f,0.f,0.f};
      #pragma unroll
      for (int ks = 0; ks < H_NB / 4; ++ks)
        acc = wmma4(axv[ks], ayv[ks], bxv[ks], byv[ks], acc);
      #pragma unroll
      for (int j = 0; j < 8; ++j) {
        const int row = j + ((lane >> 4) << 3);
        const int kr  = tile * 16 + row;
        if (kr < K_TOT)
          s_w[qg * SW_STRIDE + kr * C_IN + csub * 16 + (lane & 15)] = acc[j];
      }
    }
    __syncthreads();
  }

  if (tid < QT) {
    const int c = s_cnt[tid];
    s_nn[tid] = 1.0f / (float)((c > 1) ? c : 1);
  }
  __syncthreads();

  // ---- step 2: out_b[16,64] = weighted[16, Kb*C] @ Wb[Kb*C, 64] ----
  // 8 waves = (branch 0/1) x (d-subtile 0..3); long-K f32 WMMA GEMM
  const int branch = wave >> 2;
  const int dsub   = wave & 3;
  const int kcbase = branch ? K_MINI * C_IN : 0;
  const int kclen  = branch ? K_MID * C_IN : K_MINI * C_IN;  // 1664 / 896
  const float* W   = branch ? w_mid : w_mini;   // [(k*C+c), 64] row-major
  const int dcol   = dsub * 16 + (lane & 15);
  const int koff   = (lane >> 4) * 2;
  __builtin_prefetch(W + dcol, 0, 3);           // global_prefetch_b8 (L2 warm)
  const float* aw = s_w + (lane & 15) * SW_STRIDE + kcbase;
  v8f acc = {0.f,0.f,0.f,0.f,0.f,0.f,0.f,0.f};
  #pragma unroll 1
  for (int kc = 0; kc < kclen; kc += 16) {      // kclen % 16 == 0
    float ax[4], ay[4], bx[4], by[4];
    #pragma unroll
    for (int u = 0; u < 4; ++u) {
      const int k4 = kc + u * 4;
      ax[u] = aw[k4 + koff];
      ay[u] = aw[k4 + koff + 1];
      const float* wb = W + (size_t)(kcbase + k4 + koff) * D_OUT + dcol;
      bx[u] = wb[0];
      by[u] = wb[D_OUT];
    }
    #pragma unroll
    for (int u = 0; u < 4; ++u) acc = wmma4(ax[u], ay[u], bx[u], by[u], acc);
  }

  // density norm, store raw branch output, accumulate segment-norm stats
  float* outp = branch ? x2raw : x1raw;
  float s0 = 0.f, q0 = 0.f, s1 = 0.f, q1 = 0.f;
  #pragma unroll
  for (int j = 0; j < 8; ++j) {
    const int row = j + ((lane >> 4) << 3);
    const int qi  = qbase + row;
    const float v = acc[j] * s_nn[row];
    outp[(size_t)qi * D_OUT + dcol] = v;
    if (qi < len0) { s0 += v; q0 += v * v; }
    else           { s1 += v; q1 += v * v; }
  }
  s0 += __shfl_down(s0, 16, 32); q0 += __shfl_down(q0, 16, 32);
  s1 += __shfl_down(s1, 16, 32); q1 += __shfl_down(q1, 16, 32);
  if (lane < 16) {   // segsum layout: [branch][seg][sum|sumsq][64]
    float* sb = segsum + branch * 256;
    atomicAdd(&sb[dcol], s0);
    atomicAdd(&sb[64 + dcol], q0);
    atomicAdd(&sb[128 + dcol], s1);
    atomicAdd(&sb[128 + 64 + dcol], q1);
  }
}

// ---------------------------------------------------------------------------
// K2: finalize segment-norm stats -> mean, invstd per [branch][seg][ch]
// ---------------------------------------------------------------------------
__global__ void k2_seg_finalize(const float* __restrict__ segsum,
                                const int* __restrict__ stacklen,
                                float* __restrict__ segnorm) {
  const int t = threadIdx.x;              // 256 = 2 branch * 2 seg * 64 ch
  const int branch = t >> 7, seg = (t >> 6) & 1, ch = t & 63;
  float cnt = (float)(seg ? (N_Q - stacklen[0]) : stacklen[0]);
  cnt = fmaxf(cnt, 1.0f);
  const float s  = segsum[branch * 256 + seg * 128 + ch];
  const float sq = segsum[branch * 256 + seg * 128 + 64 + ch];
  const float mean = s / cnt;
  const float var  = sq / cnt - mean * mean;
  segnorm[branch * 256 + seg * 128 + ch]      = mean;
  segnorm[branch * 256 + seg * 128 + 64 + ch] = rsqrtf(var + EPSN);
}

// ---------------------------------------------------------------------------
// K3: segment norm apply + (x1n+x2n)@Wmm^T + concat + @Wfinal^T + BN stats
// ---------------------------------------------------------------------------
__global__ __launch_bounds__(256, 2)
void k3_head(const float* __restrict__ x1raw,
             const float* __restrict__ x2raw,
             const float* __restrict__ segnorm,
             const int*   __restrict__ stacklen,
             const float* __restrict__ w_midmini,
             const float* __restrict__ w_final,
             float* __restrict__ outpre,
             float* __restrict__ bnsum) {
  __shared__ float s_xc[QT][C_IN + 4];    // [x1n | x2p], +4 pad vs banks
  __shared__ float s_sum[QT][D_OUT + 4];  // x1n + x2n
  const int tid   = threadIdx.x;
  const int lane  = tid & 31;
  const int wave  = tid >> 5;
  const int qbase = blockIdx.x * QT;
  const int len0  = stacklen[0];

  for (int i = tid; i < QT * D_OUT; i += 256) {
    const int r = i >> 6, ch = i & 63;
    const int qi = qbase + r;
    const int seg = (qi >= len0);
    const float m1  = segnorm[seg * 128 + ch];
    const float is1 = segnorm[seg * 128 + 64 + ch];
    const float m2  = segnorm[256 + seg * 128 + ch];
    const float is2 = segnorm[256 + seg * 128 + 64 + ch];
    const float x1n = (x1raw[(size_t)qi * D_OUT + ch] - m1) * is1;
    const float x2n = (x2raw[(size_t)qi * D_OUT + ch] - m2) * is2;
    s_xc[r][ch]  = x1n;
    s_sum[r][ch] = x1n + x2n;
  }
  __syncthreads();

  // GEMM1: x2p[16,64] = s_sum @ w_midmini^T   (waves 0..3)
  if (wave < 4) {
    const int dcol = wave * 16 + (lane & 15);
    const int koff = (lane >> 4) * 2;
    v8f acc = {0.f,0.f,0.f,0.f,0.f,0.f,0.f,0.f};
    #pragma unroll
    for (int kc = 0; kc < D_OUT; kc += 4) {
      const float ax = s_sum[lane & 15][kc + koff];
      const float ay = s_sum[lane & 15][kc + koff + 1];
      const float bx = w_midmini[(size_t)dcol * D_OUT + kc + koff];
      const float by = w_midmini[(size_t)dcol * D_OUT + kc + koff + 1];
      acc = wmma4(ax, ay, bx, by, acc);
    }
    #pragma unroll
    for (int j = 0; j < 8; ++j) {
      const int row = j + ((lane >> 4) << 3);
      s_xc[row][D_OUT + dcol] = acc[j];
    }
  }
  __syncthreads();

  // GEMM2: out[16,128] = s_xc @ w_final^T   (8 waves)
  {
    const int dcol = wave * 16 + (lane & 15);
    const int koff = (lane >> 4) * 2;
    v8f acc = {0.f,0.f,0.f,0.f,0.f,0.f,0.f,0.f};
    #pragma unroll
    for (int kc = 0; kc < C_IN; kc += 4) {
      const float ax = s_xc[lane & 15][kc + koff];
      const float ay = s_xc[lane & 15][kc + koff + 1];
      const float bx = w_final[(size_t)dcol * C_IN + kc + koff];
      const float by = w_final[(size_t)dcol * C_IN + kc + koff + 1];
      acc = wmma4(ax, ay, bx, by, acc);
    }
    float s = 0.f, q = 0.f;
    #pragma unroll
    for (int j = 0; j < 8; ++j) {
      const int row = j + ((lane >> 4) << 3);
      const int qi  = qbase + row;
      const float v = acc[j];
      outpre[(size_t)qi * C_IN + dcol] = v;
      s += v; q += v * v;
    }
    s += __shfl_down(s, 16, 32); q += __shfl_down(q, 16, 32);
    if (lane < 16) {
      atomicAdd(&bnsum[dcol], s);
      atomicAdd(&bnsum[C_IN + dcol], q);
    }
  }
}

// ---------------------------------------------------------------------------
// K4: batchnorm finalize -> per-channel scale/shift
// ---------------------------------------------------------------------------
__global__ void k4_bn_finalize(const float* __restrict__ bnsum,
                               const float* __restrict__ gamma,
                               const float* __restrict__ beta,
                               float* __restrict__ bnsc) {
  const int c = threadIdx.x;              // 128
  const float mean = bnsum[c] * (1.0f / (float)N_Q);
  const float var  = bnsum[C_IN + c] * (1.0f / (float)N_Q) - mean * mean;
  const float sc   = gamma[c] * rsqrtf(var + EPSN);
  bnsc[c]        = sc;
  bnsc[C_IN + c] = beta[c] - mean * sc;
}

// ---------------------------------------------------------------------------
// K5: apply BN + relu
// ---------------------------------------------------------------------------
__global__ void k5_apply(const float* __restrict__ outpre,
                         const float* __restrict__ bnsc,
                         float* __restrict__ out) {
  const int i = blockIdx.x * 256 + threadIdx.x;
  if (i < N_Q * C_IN) {
    const int c = i & (C_IN - 1);
    const float v = outpre[i] * bnsc[c] + bnsc[C_IN + c];
    out[i] = fmaxf(v, 0.0f);
  }
}

// ---------------------------------------------------------------------------
extern "C" void kernel_launch(void* const* d_in, const int* in_sizes, int n_in,
                              void* d_out, int out_size, void* d_ws,
                              size_t ws_size, hipStream_t stream) {
  (void)in_sizes; (void)n_in; (void)out_size; (void)ws_size;
  const float* q_pts     = (const float*)d_in[0];
  const float* s_pts     = (const float*)d_in[1];
  const int*   nbi       = (const int*)d_in[2];
  const float* x         = (const float*)d_in[3];
  const int*   stacklen  = (const int*)d_in[4];
  const float* kp_mini   = (const float*)d_in[5];
  const float* w_mini    = (const float*)d_in[6];
  const float* kp_mid    = (const float*)d_in[7];
  const float* w_mid     = (const float*)d_in[8];
  const float* w_midmini = (const float*)d_in[9];
  const float* w_final   = (const float*)d_in[10];
  const float* gammaf    = (const float*)d_in[11];
  const float* betaf     = (const float*)d_in[12];
  float*       out       = (float*)d_out;

  // workspace carve (floats): ~20.5 MB total
  float* ws      = (float*)d_ws;
  float* x1raw   = ws;                        // N*64
  float* x2raw   = x1raw + (size_t)N_Q * D_OUT;
  float* outpre  = x2raw + (size_t)N_Q * D_OUT;   // N*128
  float* segsum  = outpre + (size_t)N_Q * C_IN;   // 512 (atomic accum)
  float* bnsum   = segsum + 512;                  // 256 (atomic accum)
  float* segnorm = bnsum + 256;                   // 512
  float* bnsc    = segnorm + 512;                 // 256

  // zero the atomic accumulators (contiguous: segsum+bnsum)
  hipMemsetAsync(segsum, 0, (512 + 256) * sizeof(float), stream);

  k1_kpconv_fused<<<NBLK, 256, SMEM1, stream>>>(
      q_pts, s_pts, nbi, x, stacklen, kp_mini, w_mini, kp_mid, w_mid,
      x1raw, x2raw, segsum);
  k2_seg_finalize<<<1, 256, 0, stream>>>(segsum, stacklen, segnorm);
  k3_head<<<NBLK, 256, 0, stream>>>(x1raw, x2raw, segnorm, stacklen,
                                    w_midmini, w_final, outpre, bnsum);
  k4_bn_finalize<<<1, C_IN, 0, stream>>>(bnsum, gammaf, betaf, bnsc);
  k5_apply<<<(N_Q * C_IN + 255) / 256, 256, 0, stream>>>(outpre, bnsc, out);
}